// HeteroGNN_30648886624414
// MI455X (gfx1250) — compile-verified
//
#include <hip/hip_runtime.h>

#define HD 128
#define EPSV 1e-5f

typedef __attribute__((ext_vector_type(2))) float v2f;
typedef __attribute__((ext_vector_type(8))) float v8f;

enum { GF_RELU = 1, GF_BN = 2, GF_ADD = 4 };

// ---------------------------------------------------------------------------
// GEMM: C[N,128] = epi(A[N,128] @ W[128,128] (+bias)(+addsrc)(BN)(ReLU))
// One wave = one 16x16 output tile, K=128 via 32x V_WMMA_F32_16X16X4_F32.
// Block = 256 threads = 8 waves covering all 8 column tiles of 16 rows.
// A-operand layout (ISA 7.12.2, 32-bit A 16x4): lane&15 = M, lane>>4 selects
// K pair {0,1} / {2,3}. B mirrored (lane&15 = N). C/D: VGPR r -> row r (+8 for
// upper lane half), lane&15 = column  ->  each lane owns ONE output column.
// ---------------------------------------------------------------------------
__global__ __launch_bounds__(256) void gemm128_wmma(
    const float* __restrict__ A, const float* __restrict__ W,
    const float* __restrict__ bias, const float* __restrict__ bn,
    const float* __restrict__ addsrc, float* __restrict__ C,
    int N, int flags)
{
  const int wave = threadIdx.x >> 5;
  const int lane = threadIdx.x & 31;
  const int row0 = blockIdx.x << 4;
  const int col0 = wave << 4;
  const int m    = lane & 15;
  const int kh   = lane >> 4;       // 0/1 -> K offset 0/2 inside each K=4 step
  const int bcol = col0 + m;        // single output column for this lane
  v8f acc = {};
#if defined(__HIP_DEVICE_COMPILE__) && __has_builtin(__builtin_amdgcn_wmma_f32_16x16x4_f32)
  const float* Ap = A + (size_t)(row0 + m) * HD + 2 * kh;
  const float* Bp = W + (size_t)(2 * kh) * HD + bcol;
  #pragma unroll
  for (int k = 0; k < HD; k += 4) {
    v2f a = *(const v2f*)(Ap + k);                 // A[m, k+2kh .. +1]
    v2f b;
    b.x = Bp[(size_t)k * HD];                      // B[k+2kh,   bcol]
    b.y = Bp[(size_t)(k + 1) * HD];                // B[k+2kh+1, bcol]
    acc = __builtin_amdgcn_wmma_f32_16x16x4_f32(
        false, a, false, b, (short)0, acc, false, false);
  }
#else
  // correctness fallback (no v_wmma): same acc->element mapping
  for (int k = 0; k < HD; ++k) {
    float bk = W[(size_t)k * HD + bcol];
    #pragma unroll
    for (int r = 0; r < 8; ++r)
      acc[r] = fmaf(A[(size_t)(row0 + r + 8 * kh) * HD + k], bk, acc[r]);
  }
#endif
  float bi = bias ? bias[bcol] : 0.0f;
  float g = 1.f, bb = 0.f, mu = 0.f, iv = 1.f;
  if (flags & GF_BN) {
    g  = bn[bcol];
    bb = bn[HD + bcol];
    mu = bn[2 * HD + bcol];
    iv = rsqrtf(bn[3 * HD + bcol] + EPSV);
  }
  #pragma unroll
  for (int r = 0; r < 8; ++r) {
    int row = row0 + r + 8 * kh;
    float v = acc[r] + bi;
    if (flags & GF_ADD)  v += addsrc[(size_t)row * HD + bcol];
    if (flags & GF_BN)   v = (v - mu) * iv * g + bb;
    if (flags & GF_RELU) v = fmaxf(v, 0.0f);
    C[(size_t)row * HD + bcol] = v;
  }
}

// ---- s[row] = dot(X[row,:], v[:]) ; one wave per row, 8 rows per block -----
__global__ __launch_bounds__(256) void gemv128(
    const float* __restrict__ X, const float* __restrict__ v,
    float* __restrict__ out, int N)
{
  int row  = blockIdx.x * 8 + (threadIdx.x >> 5);
  int lane = threadIdx.x & 31;
  if (row >= N) return;
  const float* xr = X + (size_t)row * HD;
  float s = xr[lane]      * v[lane]
          + xr[lane + 32] * v[lane + 32]
          + xr[lane + 64] * v[lane + 64]
          + xr[lane + 96] * v[lane + 96];
  for (int off = 16; off > 0; off >>= 1) s += __shfl_xor(s, off);
  if (lane == 0) out[row] = s;
}

// ---- wa[k] = dot(W[k,:], a[:])  (128x128 @ 128) : single block ------------
__global__ void matvec128(const float* __restrict__ W,
                          const float* __restrict__ a, float* __restrict__ wa)
{
  int k = threadIdx.x;
  float s = 0.f;
  for (int j = 0; j < HD; ++j) s = fmaf(W[(size_t)k * HD + j], a[j], s);
  wa[k] = s;
}

__global__ void fill_f32(float* p, float v, int n)
{
  int i = blockIdx.x * 256 + threadIdx.x;
  if (i < n) p[i] = v;
}

// acc[idx[i]] += 1  (degree counting / batch counts)
__global__ void add_count(const int* __restrict__ idx, float* __restrict__ acc, int n)
{
  int i = blockIdx.x * 256 + threadIdx.x;
  if (i < n) atomicAdd(&acc[idx[i]], 1.0f);
}

// o = xw/deg + gcn_b + gat_b   (self-loop term + both biases pre-folded)
__global__ void init_gcn_out(const float* __restrict__ xw, const float* __restrict__ deg,
                             const float* __restrict__ gcnb, const float* __restrict__ gatb,
                             float* __restrict__ o, int total)
{
  int i = blockIdx.x * 256 + threadIdx.x;
  if (i >= total) return;
  int node = i >> 7, k = i & 127;
  o[i] = xw[i] / deg[node] + gcnb[k] + gatb[k];
}

// GCN scatter: o[dst] += rsqrt(deg[src])*rsqrt(deg[dst]) * xw[src]
// one wave per edge, float4 per lane (128 floats)
__global__ __launch_bounds__(256) void gcn_agg(
    const int* __restrict__ src, const int* __restrict__ dst,
    const float* __restrict__ xw, const float* __restrict__ deg,
    float* __restrict__ o, int E)
{
  int e = blockIdx.x * 8 + (threadIdx.x >> 5);
  if (e >= E) return;
  int lane = threadIdx.x & 31;
  int s = src[e], d = dst[e];
  float norm = rsqrtf(deg[s]) * rsqrtf(deg[d]);
  const float4 v = *(const float4*)(xw + (size_t)s * HD + lane * 4);
  float* op = o + (size_t)d * HD + lane * 4;
  atomicAdd(op + 0, norm * v.x);
  atomicAdd(op + 1, norm * v.y);
  atomicAdd(op + 2, norm * v.z);
  atomicAdd(op + 3, norm * v.w);
}

__device__ __forceinline__ void atomicMaxF(float* addr, float val)
{
  if (val >= 0.0f) atomicMax((int*)addr, __float_as_int(val));
  else             atomicMin((unsigned int*)addr, (unsigned int)__float_as_int(val));
}

// logit = leaky_relu(s_src[src]+s_dst[dst], 0.2); m[dst] = max
__global__ void gat_logit(const int* __restrict__ src, const int* __restrict__ dst,
                          const float* __restrict__ ssrc, const float* __restrict__ sdst,
                          float* __restrict__ logit, float* __restrict__ m, int E)
{
  int e = blockIdx.x * 256 + threadIdx.x;
  if (e >= E) return;
  float v = ssrc[src[e]] + sdst[dst[e]];
  v = v > 0.0f ? v : 0.2f * v;
  logit[e] = v;
  atomicMaxF(&m[dst[e]], v);
}

// ex = exp(logit - m[dst]); den[dst] += ex  (ex overwrites logit buffer)
__global__ void gat_exp(const int* __restrict__ dst, float* __restrict__ lex,
                        const float* __restrict__ m, float* __restrict__ den, int E)
{
  int e = blockIdx.x * 256 + threadIdx.x;
  if (e >= E) return;
  int d = dst[e];
  float ex = expf(lex[e] - m[d]);
  lex[e] = ex;
  atomicAdd(&den[d], ex);
}

// o[dst] += (ex/max(den[dst],1e-16)) * hs[src]
__global__ __launch_bounds__(256) void gat_agg(
    const int* __restrict__ src, const int* __restrict__ dst,
    const float* __restrict__ ex, const float* __restrict__ den,
    const float* __restrict__ hs, float* __restrict__ o, int E)
{
  int e = blockIdx.x * 8 + (threadIdx.x >> 5);
  if (e >= E) return;
  int lane = threadIdx.x & 31;
  int s = src[e], d = dst[e];
  float alpha = ex[e] / fmaxf(den[d], 1e-16f);
  const float4 v = *(const float4*)(hs + (size_t)s * HD + lane * 4);
  float* op = o + (size_t)d * HD + lane * 4;
  atomicAdd(op + 0, alpha * v.x);
  atomicAdd(op + 1, alpha * v.y);
  atomicAdd(op + 2, alpha * v.z);
  atomicAdd(op + 3, alpha * v.w);
}

// per-block LDS partial sums (batch is sorted -> low contention), then global
__global__ __launch_bounds__(256) void pool_accum(
    const float* __restrict__ h, const int* __restrict__ batch,
    float* __restrict__ gsum, int N)
{
  __shared__ float acc[16 * HD];
  int tid = threadIdx.x;
  for (int i = tid; i < 16 * HD; i += 256) acc[i] = 0.0f;
  __syncthreads();
  int col = tid & 127, half = tid >> 7;
  int n0 = blockIdx.x * 512;
  int nEnd = n0 + 512; if (nEnd > N) nEnd = N;
  for (int n = n0 + half; n < nEnd; n += 2)
    atomicAdd(&acc[batch[n] * HD + col], h[(size_t)n * HD + col]);
  __syncthreads();
  for (int i = tid; i < 16 * HD; i += 256)
    if (acc[i] != 0.0f) atomicAdd(&gsum[i], acc[i]);
}

// mean + concat + Linear(256->128)+ReLU + Linear(128->2); tiny -> one block
__global__ __launch_bounds__(256) void mlp_head(
    const float* __restrict__ gsum1, const float* __restrict__ cnt1,
    const float* __restrict__ gsum2, const float* __restrict__ cnt2,
    const float* __restrict__ w1, const float* __restrict__ b1,
    const float* __restrict__ w2, const float* __restrict__ b2,
    float* __restrict__ out)
{
  __shared__ float r[16 * 256];
  __shared__ float t[16 * 128];
  int tid = threadIdx.x;
  for (int i = tid; i < 16 * 256; i += 256) {
    int b = i >> 8, k = i & 255;
    float v = (k < 128) ? gsum1[b * 128 + k] / fmaxf(cnt1[b], 1.0f)
                        : gsum2[b * 128 + (k - 128)] / fmaxf(cnt2[b], 1.0f);
    r[i] = v;
  }
  __syncthreads();
  for (int i = tid; i < 16 * 128; i += 256) {
    int b = i >> 7, j = i & 127;
    float s = b1[j];
    for (int k = 0; k < 256; ++k) s = fmaf(r[b * 256 + k], w1[k * 128 + j], s);
    t[i] = fmaxf(s, 0.0f);
  }
  __syncthreads();
  if (tid < 32) {
    int b = tid >> 1, o = tid & 1;
    float s = b2[o];
    for (int j = 0; j < 128; ++j) s = fmaf(t[b * 128 + j], w2[j * 2 + o], s);
    out[b * 2 + o] = s;
  }
}

// ---------------------------------------------------------------------------
extern "C" void kernel_launch(void* const* d_in, const int* in_sizes, int n_in,
                              void* d_out, int out_size, void* d_ws, size_t ws_size,
                              hipStream_t stream)
{
  const float* x1      = (const float*)d_in[0];
  const float* x2      = (const float*)d_in[1];
  const float* pre_w1  = (const float*)d_in[2];
  const float* pre_b1  = (const float*)d_in[3];
  const float* pre_w2  = (const float*)d_in[4];
  const float* pre_b2  = (const float*)d_in[5];
  const float* bn_pre1 = (const float*)d_in[6];
  const float* bn_pre2 = (const float*)d_in[7];
  const float* bn_post1= (const float*)d_in[8];
  const float* bn_post2= (const float*)d_in[9];
  const float* gcn1_w  = (const float*)d_in[10];
  const float* gcn1_b  = (const float*)d_in[11];
  const float* gcn2_w  = (const float*)d_in[12];
  const float* gcn2_b  = (const float*)d_in[13];
  const float* gat12_ws= (const float*)d_in[14];
  const float* gat12_wd= (const float*)d_in[15];
  const float* gat12_as= (const float*)d_in[16];
  const float* gat12_ad= (const float*)d_in[17];
  const float* gat12_b = (const float*)d_in[18];
  const float* gat21_ws= (const float*)d_in[19];
  const float* gat21_wd= (const float*)d_in[20];
  const float* gat21_as= (const float*)d_in[21];
  const float* gat21_ad= (const float*)d_in[22];
  const float* gat21_b = (const float*)d_in[23];
  const float* skip1_w = (const float*)d_in[24];
  const float* skip1_b = (const float*)d_in[25];
  const float* skip2_w = (const float*)d_in[26];
  const float* skip2_b = (const float*)d_in[27];
  const float* post_w1 = (const float*)d_in[28];
  const float* post_b1 = (const float*)d_in[29];
  const float* post_w2 = (const float*)d_in[30];
  const float* post_b2 = (const float*)d_in[31];
  const float* lin1_w  = (const float*)d_in[32];
  const float* lin1_b  = (const float*)d_in[33];
  const float* lin2_w  = (const float*)d_in[34];
  const float* lin2_b  = (const float*)d_in[35];
  const int*   e11     = (const int*)d_in[36];
  const int*   e22     = (const int*)d_in[37];
  const int*   e12     = (const int*)d_in[38];
  const int*   e21     = (const int*)d_in[39];
  const int*   batch1  = (const int*)d_in[40];
  const int*   batch2  = (const int*)d_in[41];

  const int N1 = in_sizes[40];
  const int N2 = in_sizes[41];
  const int E  = in_sizes[36] / 2;
  const int Nmax = N1 > N2 ? N1 : N2;
  const int L  = 2;

  const int* e11s = e11;       const int* e11d = e11 + E;
  const int* e22s = e22;       const int* e22d = e22 + E;
  const int* e12s = e12;       const int* e12d = e12 + E;
  const int* e21s = e21;       const int* e21d = e21 + E;

  // ---- workspace carve-up (float units, 256B aligned chunks) ----
  float* p = (float*)d_ws;
  auto take = [&](size_t nElem) {
    float* q = p;
    p += ((nElem + 63) & ~(size_t)63);
    return q;
  };
  float* h1   = take((size_t)N1 * HD);
  float* h2   = take((size_t)N2 * HD);
  float* o1   = take((size_t)N1 * HD);
  float* o2   = take((size_t)N2 * HD);
  float* t    = take((size_t)Nmax * HD);   // xw / hs staging (reused serially)
  float* deg1 = take(N1);
  float* deg2 = take(N2);
  float* sv1  = take(Nmax);                // s_src
  float* sv2  = take(Nmax);                // s_dst
  float* mbuf = take(Nmax);                // segment max
  float* den  = take(Nmax);                // segment sum of exp
  float* exbf = take(E);                   // per-edge logit/exp
  float* wa   = take(HD);                  // Wd @ a_d
  float* gs1  = take(16 * HD);
  float* gs2  = take(16 * HD);
  float* cnt1 = take(16);
  float* cnt2 = take(16);
  (void)ws_size; (void)n_in; (void)out_size;

  const dim3 blk(256);
  auto gE1  = dim3((E + 255) / 256);       // one thread / edge
  auto gE8  = dim3((E + 7) / 8);           // one wave / edge
  auto gN1t = dim3((N1 * HD + 255) / 256);
  auto gN2t = dim3((N2 * HD + 255) / 256);

  // ---- degrees (constant across layers) ----
  fill_f32<<<dim3((N1 + 255) / 256), blk, 0, stream>>>(deg1, 1.0f, N1);
  fill_f32<<<dim3((N2 + 255) / 256), blk, 0, stream>>>(deg2, 1.0f, N2);
  add_count<<<gE1, blk, 0, stream>>>(e11d, deg1, E);
  add_count<<<gE1, blk, 0, stream>>>(e22d, deg2, E);

  // ---- pre-lin + BN + ReLU ----
  gemm128_wmma<<<dim3(N1 / 16), blk, 0, stream>>>(x1, pre_w1, pre_b1, bn_pre1, nullptr, h1, N1, GF_BN | GF_RELU);
  gemm128_wmma<<<dim3(N2 / 16), blk, 0, stream>>>(x2, pre_w2, pre_b2, bn_pre2, nullptr, h2, N2, GF_BN | GF_RELU);

  for (int i = 0; i < L; ++i) {
    const size_t WO = (size_t)i * HD * HD, BO = (size_t)i * HD;

    // ===== o1 = GCN(h1; e11) + GAT(h2 -> h1; e21) =====
    gemm128_wmma<<<dim3(N1 / 16), blk, 0, stream>>>(h1, gcn1_w + WO, nullptr, nullptr, nullptr, t, N1, 0);
    init_gcn_out<<<gN1t, blk, 0, stream>>>(t, deg1, gcn1_b + BO, gat21_b + BO, o1, N1 * HD);
    gcn_agg<<<gE8, blk, 0, stream>>>(e11s, e11d, t, deg1, o1, E);

    gemm128_wmma<<<dim3(N2 / 16), blk, 0, stream>>>(h2, gat21_ws + WO, nullptr, nullptr, nullptr, t, N2, 0); // hs
    gemv128<<<dim3((N2 + 7) / 8), blk, 0, stream>>>(t, gat21_as + BO, sv1, N2);       // s_src
    matvec128<<<dim3(1), dim3(128), 0, stream>>>(gat21_wd + WO, gat21_ad + BO, wa);
    gemv128<<<dim3((N1 + 7) / 8), blk, 0, stream>>>(h1, wa, sv2, N1);                 // s_dst
    fill_f32<<<dim3((N1 + 255) / 256), blk, 0, stream>>>(mbuf, -3.0e38f, N1);
    fill_f32<<<dim3((N1 + 255) / 256), blk, 0, stream>>>(den, 0.0f, N1);
    gat_logit<<<gE1, blk, 0, stream>>>(e21s, e21d, sv1, sv2, exbf, mbuf, E);
    gat_exp<<<gE1, blk, 0, stream>>>(e21d, exbf, mbuf, den, E);
    gat_agg<<<gE8, blk, 0, stream>>>(e21s, e21d, exbf, den, t, o1, E);

    // ===== o2 = GCN(h2; e22) + GAT(h1 -> h2; e12) =====
    gemm128_wmma<<<dim3(N2 / 16), blk, 0, stream>>>(h2, gcn2_w + WO, nullptr, nullptr, nullptr, t, N2, 0);
    init_gcn_out<<<gN2t, blk, 0, stream>>>(t, deg2, gcn2_b + BO, gat12_b + BO, o2, N2 * HD);
    gcn_agg<<<gE8, blk, 0, stream>>>(e22s, e22d, t, deg2, o2, E);

    gemm128_wmma<<<dim3(N1 / 16), blk, 0, stream>>>(h1, gat12_ws + WO, nullptr, nullptr, nullptr, t, N1, 0); // hs
    gemv128<<<dim3((N1 + 7) / 8), blk, 0, stream>>>(t, gat12_as + BO, sv1, N1);       // s_src
    matvec128<<<dim3(1), dim3(128), 0, stream>>>(gat12_wd + WO, gat12_ad + BO, wa);
    gemv128<<<dim3((N2 + 7) / 8), blk, 0, stream>>>(h2, wa, sv2, N2);                 // s_dst
    fill_f32<<<dim3((N2 + 255) / 256), blk, 0, stream>>>(mbuf, -3.0e38f, N2);
    fill_f32<<<dim3((N2 + 255) / 256), blk, 0, stream>>>(den, 0.0f, N2);
    gat_logit<<<gE1, blk, 0, stream>>>(e12s, e12d, sv1, sv2, exbf, mbuf, E);
    gat_exp<<<gE1, blk, 0, stream>>>(e12d, exbf, mbuf, den, E);
    gat_agg<<<gE8, blk, 0, stream>>>(e12s, e12d, exbf, den, t, o2, E);

    // ===== skip: h = relu(o + o@W + b) =====
    gemm128_wmma<<<dim3(N1 / 16), blk, 0, stream>>>(o1, skip1_w + WO, skip1_b + BO, nullptr, o1, h1, N1, GF_ADD | GF_RELU);
    gemm128_wmma<<<dim3(N2 / 16), blk, 0, stream>>>(o2, skip2_w + WO, skip2_b + BO, nullptr, o2, h2, N2, GF_ADD | GF_RELU);
  }

  // ---- post-lin + BN ----
  gemm128_wmma<<<dim3(N1 / 16), blk, 0, stream>>>(h1, post_w1, post_b1, bn_post1, nullptr, o1, N1, GF_BN);
  gemm128_wmma<<<dim3(N2 / 16), blk, 0, stream>>>(h2, post_w2, post_b2, bn_post2, nullptr, o2, N2, GF_BN);

  // ---- mean pool per batch + head MLP ----
  fill_f32<<<dim3(8), blk, 0, stream>>>(gs1, 0.0f, 16 * HD);
  fill_f32<<<dim3(8), blk, 0, stream>>>(gs2, 0.0f, 16 * HD);
  fill_f32<<<dim3(1), blk, 0, stream>>>(cnt1, 0.0f, 16);
  fill_f32<<<dim3(1), blk, 0, stream>>>(cnt2, 0.0f, 16);
  pool_accum<<<dim3((N1 + 511) / 512), blk, 0, stream>>>(o1, batch1, gs1, N1);
  pool_accum<<<dim3((N2 + 511) / 512), blk, 0, stream>>>(o2, batch2, gs2, N2);
  add_count<<<dim3((N1 + 255) / 256), blk, 0, stream>>>(batch1, cnt1, N1);
  add_count<<<dim3((N2 + 255) / 256), blk, 0, stream>>>(batch2, cnt2, N2);

  mlp_head<<<dim3(1), blk, 0, stream>>>(gs1, cnt1, gs2, cnt2,
                                        lin1_w, lin1_b, lin2_w, lin2_b,
                                        (float*)d_out);
}